// GO_to_P_9457517986563
// MI455X (gfx1250) — compile-verified
//
#include <hip/hip_runtime.h>

typedef __attribute__((ext_vector_type(2))) float v2f;
typedef __attribute__((ext_vector_type(8))) float v8f;

#define D_DIM 128

// ---------------------------------------------------------------------------
// 1) zero the workspace accumulators (float4 stores, graph-capture safe)
// ---------------------------------------------------------------------------
__global__ void zero_f4_kernel(float4* __restrict__ p, long n4) {
    long i = (long)blockIdx.x * blockDim.x + threadIdx.x;
    if (i < n4) p[i] = make_float4(0.f, 0.f, 0.f, 0.f);
}

// ---------------------------------------------------------------------------
// 2) scatter-add: one wave32 per edge, each lane handles 4 features.
//    gridDim.y in {0,1,2} selects the relation. Hardware fp32 atomics.
// ---------------------------------------------------------------------------
__global__ void scatter_add_kernel(
    const float* __restrict__ h_mf, const float* __restrict__ h_bp,
    const float* __restrict__ h_cc,
    const int* __restrict__ mf_src, const int* __restrict__ mf_dst,
    const int* __restrict__ bp_src, const int* __restrict__ bp_dst,
    const int* __restrict__ cc_src, const int* __restrict__ cc_dst,
    float* __restrict__ agg_mf, float* __restrict__ agg_bp,
    float* __restrict__ agg_cc, int E)
{
    const float* h; const int* src; const int* dst; float* agg;
    switch (blockIdx.y) {
        case 0:  h = h_mf; src = mf_src; dst = mf_dst; agg = agg_mf; break;
        case 1:  h = h_bp; src = bp_src; dst = bp_dst; agg = agg_bp; break;
        default: h = h_cc; src = cc_src; dst = cc_dst; agg = agg_cc; break;
    }
    const int t    = blockIdx.x * blockDim.x + threadIdx.x;
    const int edge = t >> 5;       // wave32: one wave per edge
    const int lane = t & 31;       // lane covers 4 consecutive features
    if (edge >= E) return;

    const int s = src[edge];
    const int d = dst[edge];
    const float4 v = *(const float4*)(h + (long)s * D_DIM + lane * 4);
    float* o = agg + (long)d * D_DIM + lane * 4;
    // lowers to global_atomic_add_f32 (no CAS loop)
    unsafeAtomicAdd(o + 0, v.x);
    unsafeAtomicAdd(o + 1, v.y);
    unsafeAtomicAdd(o + 2, v.z);
    unsafeAtomicAdd(o + 3, v.w);
}

// ---------------------------------------------------------------------------
// 3) fused  out = relu(agg_mf@W_mf+b_mf) + relu(agg_bp@W_bp+b_bp)
//                 + relu(agg_cc@W_cc+b_cc)
//    using V_WMMA_F32_16X16X4_F32 (full fp32 precision).
//    Block: 256 threads = 8 waves. blockIdx.y picks a 16-column slice,
//    each wave owns one 16-row tile. W slices (3 x 128 x 16) staged in LDS.
// ---------------------------------------------------------------------------
__global__ __launch_bounds__(256)
void gemm_relu_sum_kernel(
    const float* __restrict__ agg_mf, const float* __restrict__ agg_bp,
    const float* __restrict__ agg_cc,
    const float* __restrict__ W_mf, const float* __restrict__ b_mf,
    const float* __restrict__ W_bp, const float* __restrict__ b_bp,
    const float* __restrict__ W_cc, const float* __restrict__ b_cc,
    float* __restrict__ out, int n_p)
{
    __shared__ float Ws[3][D_DIM][16];          // 24 KB

    const int colBase = blockIdx.y * 16;

    const float* Wg[3] = {W_mf, W_bp, W_cc};
    // Cooperative load: 3 * 2048 floats, 256 threads -> 24 floats each.
    #pragma unroll
    for (int r = 0; r < 3; ++r) {
        for (int idx = threadIdx.x; idx < D_DIM * 16; idx += 256) {
            const int k = idx >> 4;
            const int n = idx & 15;
            Ws[r][k][n] = Wg[r][k * D_DIM + colBase + n];
        }
    }
    __syncthreads();

    const int wave    = threadIdx.x >> 5;
    const int lane    = threadIdx.x & 31;
    const int rowTile = (blockIdx.x * 8 + wave) * 16;
    if (rowTile >= n_p) return;                 // wave-uniform: EXEC stays full

    const int mrow  = lane & 15;                // A layout: M = lane % 16
    const int khalf = (lane >> 4) * 2;          // A layout: lanes 16-31 hold K+2,K+3
    const int ncol  = lane & 15;                // B/C/D layout: N = lane % 16

    const float* Ag[3] = {agg_mf, agg_bp, agg_cc};
    const float* bg[3] = {b_mf, b_bp, b_cc};

    v8f outacc = {};

    #pragma unroll
    for (int r = 0; r < 3; ++r) {
        v8f c = {};
        const float* A = Ag[r] + (long)(rowTile + mrow) * D_DIM;
        #pragma unroll
        for (int k = 0; k < D_DIM; k += 4) {
            const float2 af = *(const float2*)(A + k + khalf);
            v2f a; a.x = af.x; a.y = af.y;
            v2f b; b.x = Ws[r][k + khalf][ncol];
                   b.y = Ws[r][k + khalf + 1][ncol];
            // 8 args: (neg_a, A, neg_b, B, c_mod, C, reuse_a, reuse_b)
            c = __builtin_amdgcn_wmma_f32_16x16x4_f32(
                    false, a, false, b, (short)0, c, false, false);
        }
        const float bias = bg[r][colBase + ncol];
        #pragma unroll
        for (int g = 0; g < 8; ++g)
            outacc[g] += fmaxf(c[g] + bias, 0.0f);
    }

    // C/D layout: VGPR g holds row (g) for lanes 0-15, row (g+8) for lanes 16-31
    const int rOff = (lane >> 4) * 8;
    float* o = out + (long)rowTile * D_DIM + colBase + ncol;
    #pragma unroll
    for (int g = 0; g < 8; ++g)
        o[(long)(g + rOff) * D_DIM] = outacc[g];
}

// ---------------------------------------------------------------------------
extern "C" void kernel_launch(void* const* d_in, const int* in_sizes, int n_in,
                              void* d_out, int out_size, void* d_ws, size_t ws_size,
                              hipStream_t stream) {
    // setup_inputs() order:
    // 0:h_p 1:h_mf 2:h_bp 3:h_cc 4:mf_src 5:mf_dst 6:bp_src 7:bp_dst
    // 8:cc_src 9:cc_dst 10:W_mf 11:b_mf 12:W_bp 13:b_bp 14:W_cc 15:b_cc
    const float* h_mf = (const float*)d_in[1];
    const float* h_bp = (const float*)d_in[2];
    const float* h_cc = (const float*)d_in[3];
    const int* mf_src = (const int*)d_in[4];
    const int* mf_dst = (const int*)d_in[5];
    const int* bp_src = (const int*)d_in[6];
    const int* bp_dst = (const int*)d_in[7];
    const int* cc_src = (const int*)d_in[8];
    const int* cc_dst = (const int*)d_in[9];
    const float* W_mf = (const float*)d_in[10];
    const float* b_mf = (const float*)d_in[11];
    const float* W_bp = (const float*)d_in[12];
    const float* b_bp = (const float*)d_in[13];
    const float* W_cc = (const float*)d_in[14];
    const float* b_cc = (const float*)d_in[15];

    const int n_p = in_sizes[0] / D_DIM;   // 100000
    const int E   = in_sizes[4];           // 500000

    float* agg_mf = (float*)d_ws;
    float* agg_bp = agg_mf + (size_t)n_p * D_DIM;
    float* agg_cc = agg_bp + (size_t)n_p * D_DIM;

    // 1) zero accumulators
    const long n4 = (long)3 * n_p * D_DIM / 4;
    zero_f4_kernel<<<(unsigned)((n4 + 255) / 256), 256, 0, stream>>>(
        (float4*)d_ws, n4);

    // 2) scatter-add, 3 relations via gridDim.y
    dim3 sgrid((unsigned)(((long)E * 32 + 255) / 256), 3);
    scatter_add_kernel<<<sgrid, 256, 0, stream>>>(
        h_mf, h_bp, h_cc, mf_src, mf_dst, bp_src, bp_dst, cc_src, cc_dst,
        agg_mf, agg_bp, agg_cc, E);

    // 3) fused WMMA GEMM + bias + relu + sum
    dim3 ggrid((unsigned)((n_p + 127) / 128), D_DIM / 16);
    gemm_relu_sum_kernel<<<ggrid, 256, 0, stream>>>(
        agg_mf, agg_bp, agg_cc, W_mf, b_mf, W_bp, b_bp, W_cc, b_cc,
        (float*)d_out, n_p);
}